// ClosePenalty_40381282517176
// MI455X (gfx1250) — compile-verified
//
#include <hip/hip_runtime.h>

#define TILE    1024   // edges per staged tile
#define THREADS 256    // 8 wave32 per block
#define MAXB    8      // B == 8 (out_size)
#define ROWPAD  9      // 9 floats/row -> odd stride, banks spread
#define NWAVES  (THREADS / 32)

__global__ void zero_out_kernel(float* __restrict__ out, int n) {
    int i = blockIdx.x * blockDim.x + threadIdx.x;
    if (i < n) out[i] = 0.0f;
}

// Async DMA: 16 bytes global -> LDS, tracked by ASYNCcnt (CDNA5-specific path).
__device__ __forceinline__ void async_copy_b128(const void* gptr, void* ldsptr) {
    unsigned lds          = (unsigned)(unsigned long long)ldsptr;  // low 32b of generic shared addr == LDS offset
    unsigned long long ga = (unsigned long long)gptr;
    asm volatile("global_load_async_to_lds_b128 %0, %1, off"
                 :: "v"(lds), "v"(ga) : "memory");
}

__device__ __forceinline__ void edge_term(
    int n, int i, int j, float sx, float sy, float sz,
    const float* __restrict__ pos, const int* __restrict__ elm,
    const float2* s_kr, int N, float* row)
{
    const int ai = n * N + i;
    const int aj = n * N + j;
    const float* pi = pos + ai * 3;       // L2-resident gather (384 KB table)
    const float* pj = pos + aj * 3;
    float dx = pj[0] - pi[0] + sx;
    float dy = pj[1] - pi[1] + sy;
    float dz = pj[2] - pi[2] + sz;
    float sod = dx * dx + dy * dy + dz * dz;
    float dis = __builtin_amdgcn_sqrtf(sod);      // raw v_sqrt_f32 (~1 ULP)
    int e0 = elm[ai], e1 = elm[aj];               // L2-resident gather (128 KB table)
    float2 kr0 = s_kr[e0];                        // ds_load_b64 per endpoint
    float2 kr1 = s_kr[e1];
    float k = kr0.x + kr1.x;
    float R = kr0.y + kr1.y;
    float df = dis - R;
    float ev = (dis < R) ? (k * df * df) : 0.0f;
    // Per-thread exclusive row: lowers to ds_add_f32 (no return), in program
    // order within the wave -> deterministic; 1 DS op instead of 24 VALU ops.
    atomicAdd(&row[n], ev);
}

__global__ __launch_bounds__(THREADS) void close_penalty_kernel(
    const float* __restrict__ pos, const float* __restrict__ sft,
    const float* __restrict__ ktab, const float* __restrict__ rtab,
    const int* __restrict__ elm,
    const int* __restrict__ edge_n, const int* __restrict__ edge_i,
    const int* __restrict__ edge_j,
    float* __restrict__ out, float* __restrict__ ws,
    int E, int N, int B, int nelem, int nTiles, int tailStart, int use_ws)
{
    __shared__ __align__(16) int   s_n[2][TILE];
    __shared__ __align__(16) int   s_i[2][TILE];
    __shared__ __align__(16) int   s_j[2][TILE];
    __shared__ __align__(16) float s_s[2][TILE * 3];
    __shared__ __align__(8)  float s_kr[2 * 64];        // interleaved {k, r}
    __shared__ float s_acc[THREADS][ROWPAD];            // per-thread bins (exclusive rows)
    __shared__ float s_wb[NWAVES][MAXB];                // per-wave partials

    const int t = threadIdx.x;
    if (t < nelem && t < 64) { s_kr[2 * t] = ktab[t]; s_kr[2 * t + 1] = rtab[t]; }
    float* row = &s_acc[t][0];
#pragma unroll
    for (int b = 0; b < MAXB; ++b) row[b] = 0.0f;
    __syncthreads();
    const float2* kr = reinterpret_cast<const float2*>(s_kr);

    const int grid = gridDim.x;

    // Stage one 1024-edge tile (24 KB) into LDS: 6 async b128 per wave.
    auto issue = [&](int buf, int tile) {
        const int ebase = tile * TILE + t * 4;                    // 16B-aligned
        async_copy_b128(edge_n + ebase, &s_n[buf][t * 4]);
        async_copy_b128(edge_i + ebase, &s_i[buf][t * 4]);
        async_copy_b128(edge_j + ebase, &s_j[buf][t * 4]);
        const float* g = sft + (long)tile * (TILE * 3) + t * 12;  // 48B/thread, 16B-aligned
        float* l = &s_s[buf][t * 12];
        async_copy_b128(g + 0, l + 0);
        async_copy_b128(g + 4, l + 4);
        async_copy_b128(g + 8, l + 8);
    };

    const int tile0 = blockIdx.x;
    if (tile0 < nTiles)        issue(0, tile0);
    if (tile0 + grid < nTiles) issue(1, tile0 + grid);

    int cur = 0;
    for (int tile = tile0; tile < nTiles; tile += grid, cur ^= 1) {
        const bool hasNext = (tile + grid) < nTiles;
        // Wait for this buffer's 6 async ops; keep next buffer's 6 in flight.
        if (hasNext) asm volatile("s_wait_asynccnt 0x6" ::: "memory");
        else         asm volatile("s_wait_asynccnt 0x0" ::: "memory");
        __syncthreads();   // publish LDS written by all waves' async engines

        int4 nq = *reinterpret_cast<const int4*>(&s_n[cur][t * 4]);
        int4 iq = *reinterpret_cast<const int4*>(&s_i[cur][t * 4]);
        int4 jq = *reinterpret_cast<const int4*>(&s_j[cur][t * 4]);
        const float4* sv = reinterpret_cast<const float4*>(&s_s[cur][t * 12]);
        float4 s0 = sv[0], s1 = sv[1], s2 = sv[2];

        edge_term(nq.x, iq.x, jq.x, s0.x, s0.y, s0.z, pos, elm, kr, N, row);
        edge_term(nq.y, iq.y, jq.y, s0.w, s1.x, s1.y, pos, elm, kr, N, row);
        edge_term(nq.z, iq.z, jq.z, s1.z, s1.w, s2.x, pos, elm, kr, N, row);
        edge_term(nq.w, iq.w, jq.w, s2.y, s2.z, s2.w, pos, elm, kr, N, row);

        __syncthreads();   // all reads of buffer `cur` done before refilling it
        const int pf = tile + 2 * grid;
        if (pf < nTiles) issue(cur, pf);
    }

    // Tail edges (E % TILE), direct path.
    for (int e = tailStart + blockIdx.x * THREADS + t; e < E; e += grid * THREADS) {
        edge_term(edge_n[e], edge_i[e], edge_j[e],
                  sft[e * 3 + 0], sft[e * 3 + 1], sft[e * 3 + 2],
                  pos, elm, kr, N, row);
    }

    __syncthreads();   // all ds_add_f32 done; rows stable

    // Deterministic block reduction: load own row -> wave32 shuffle tree ->
    // per-wave LDS slots -> thread b sums the 8 wave partials in fixed order.
    float acc[MAXB];
#pragma unroll
    for (int b = 0; b < MAXB; ++b) {
        float v = row[b];
        v += __shfl_down(v, 16);
        v += __shfl_down(v, 8);
        v += __shfl_down(v, 4);
        v += __shfl_down(v, 2);
        v += __shfl_down(v, 1);
        acc[b] = v;
    }
    if ((t & 31) == 0) {
        const int w = t >> 5;
#pragma unroll
        for (int b = 0; b < MAXB; ++b) s_wb[w][b] = acc[b];
    }
    __syncthreads();
    if (t < MAXB) {
        float tot = 0.0f;
#pragma unroll
        for (int w = 0; w < NWAVES; ++w) tot += s_wb[w][t];
        if (use_ws) ws[(long)blockIdx.x * MAXB + t] = tot;     // deterministic path
        else if (t < B) atomicAdd(&out[t], tot);               // fallback
    }
}

// Deterministic combine of grid x MAXB partials: fixed-stride chunk sums,
// then fixed-order chunk combine. Single block.
__global__ __launch_bounds__(THREADS) void reduce_bins_kernel(
    const float* __restrict__ ws, float* __restrict__ out, int nblocks, int B)
{
    __shared__ float s_part[32][MAXB];
    const int t = threadIdx.x;
    const int bin   = t & (MAXB - 1);
    const int chunk = t >> 3;                 // 0..31
    float s = 0.0f;
    for (int c = chunk; c < nblocks; c += 32) s += ws[(long)c * MAXB + bin];
    s_part[chunk][bin] = s;
    __syncthreads();
    if (t < MAXB) {
        float tot = 0.0f;
#pragma unroll
        for (int c = 0; c < 32; ++c) tot += s_part[c][t];
        if (t < B) out[t] = tot;
    }
}

extern "C" void kernel_launch(void* const* d_in, const int* in_sizes, int n_in,
                              void* d_out, int out_size, void* d_ws, size_t ws_size,
                              hipStream_t stream) {
    const float* pos = (const float*)d_in[0];
    const float* sft = (const float*)d_in[1];
    const float* kt  = (const float*)d_in[2];
    const float* rt  = (const float*)d_in[3];
    const int*   elm = (const int*)d_in[4];
    const int*   en  = (const int*)d_in[5];
    const int*   ei  = (const int*)d_in[6];
    const int*   ej  = (const int*)d_in[7];
    float* out = (float*)d_out;
    float* ws  = (float*)d_ws;

    const int E     = in_sizes[5];
    const int B     = out_size;                      // eng_mol is [B,1]
    const int N     = (B > 0) ? in_sizes[4] / B : 0; // elm is [B,N]
    const int nelem = in_sizes[2];                   // k_table length

    const int nTiles    = E / TILE;
    const int tailStart = nTiles * TILE;

    int grid = 1024;                                 // ~8 tiles/block -> deep async pipelining
    if (nTiles > 0 && nTiles < grid) grid = nTiles;
    if (grid < 1) grid = 1;

    const int use_ws = (ws != nullptr) &&
                       (ws_size >= (size_t)grid * MAXB * sizeof(float)) &&
                       (B <= MAXB);

    if (!use_ws) {
        hipLaunchKernelGGL(zero_out_kernel, dim3((B + 63) / 64), dim3(64), 0, stream, out, B);
    }
    hipLaunchKernelGGL(close_penalty_kernel, dim3(grid), dim3(THREADS), 0, stream,
                       pos, sft, kt, rt, elm, en, ei, ej, out, ws,
                       E, N, B, nelem, nTiles, tailStart, use_ws);
    if (use_ws) {
        hipLaunchKernelGGL(reduce_bins_kernel, dim3(1), dim3(THREADS), 0, stream,
                           ws, out, grid, B);
    }
}